// ChamferDistance_28913719836862
// MI455X (gfx1250) — compile-verified
//
#include <hip/hip_runtime.h>

typedef __attribute__((ext_vector_type(2))) float v2f;
typedef __attribute__((ext_vector_type(8))) float v8f;

#define TILE 16
#define WAVES_PER_BLOCK 8
#define BLOCK (WAVES_PER_BLOCK * 32)

// For each row p of P (16-row tile per wave), compute min_q dist(p,q) over all
// nQ points of Q with V_WMMA_F32_16X16X4_F32, then atomically accumulate the
// sum of per-row mins into *acc (one atomicAdd per wave).
//
// Homogeneous-coordinate trick using the K=4 slot:
//   A row = (-2*p0, -2*p1, -2*p2, 1)
//   B col = ( q0,    q1,    q2,  |q|^2)
//   =>  c[m][n] = -2<p_m, q_n> + |q_n|^2
//   =>  dist    = c + |p_m|^2   (single add + min in the epilogue)
__global__ __launch_bounds__(BLOCK) void chamfer_rowmin_kernel(
    const float* __restrict__ P, const float* __restrict__ Q,
    int nP, int nQ, float* __restrict__ acc)
{
  const int lane = threadIdx.x & 31;
  const int wave = threadIdx.x >> 5;
  const int tilesP = nP / TILE;
  const int t  = blockIdx.x * WAVES_PER_BLOCK + wave;  // global row-tile id
  const int b  = t / tilesP;                           // batch index
  const int rt = t % tilesP;                           // row tile within batch
  const int i0 = rt * TILE;
  const int col = lane & 15;
  const int hi  = lane >> 4;                           // 0 = lanes 0-15, 1 = 16-31

  const float* Pb = P + (size_t)b * nP * 3;
  const float* Qb = Q + (size_t)b * nQ * 3;

  // ---- A fragment: 16x4 f32, row i0+col = (-2p0, -2p1, -2p2, 1) ----
  // ISA layout: lanes 0-15 hold (M=lane, K=0) in V0 and (M=lane, K=1) in V1;
  //             lanes 16-31 hold K=2 in V0 and K=3 in V1.
  const float* ap = Pb + (size_t)(i0 + col) * 3;
  {
  }
  float p0 = ap[0], p1 = ap[1], p2 = ap[2];
  v2f A;
  A.x = hi ? -2.0f * p2 : -2.0f * p0;
  A.y = hi ? 1.0f       : -2.0f * p1;

  // Squared norms of the 8 rows this lane's C fragment covers
  // (C VGPR v -> row i0 + hi*8 + v, column j0 + col).
  float xx[8];
#pragma unroll
  for (int v = 0; v < 8; ++v) {
    const float* p = Pb + (size_t)(i0 + hi * 8 + v) * 3;
    xx[v] = p[0] * p[0] + p[1] * p[1] + p[2] * p[2];
  }

  float rmin[8];
#pragma unroll
  for (int v = 0; v < 8; ++v) rmin[v] = 3.402823466e38f;

  for (int j0 = 0; j0 < nQ; j0 += TILE) {
    // ---- B fragment: 4x16 f32, column j0+col = (q0, q1, q2, |q|^2) ----
    // Mirror layout of A: lanes 0-15 hold (K=0,N=col)/(K=1,N=col);
    // lanes 16-31 hold (K=2)/(K=3).
    const float* qp = Qb + (size_t)(j0 + col) * 3;
    float q0 = qp[0], q1 = qp[1], q2 = qp[2];
    float qq = q0 * q0 + q1 * q1 + q2 * q2;
    v2f Bv;
    Bv.x = hi ? q2 : q0;
    Bv.y = hi ? qq : q1;

    v8f c = {0.f, 0.f, 0.f, 0.f, 0.f, 0.f, 0.f, 0.f};
    // 8 args: (neg_a, A, neg_b, B, c_mod, C, reuse_a, reuse_b)
    c = __builtin_amdgcn_wmma_f32_16x16x4_f32(
        false, A, false, Bv, (short)0, c, false, false);

#pragma unroll
    for (int v = 0; v < 8; ++v)
      rmin[v] = fminf(rmin[v], c[v] + xx[v]);
  }

  // Min-reduce across the 16 lanes of each half-wave (covers all columns).
#pragma unroll
  for (int off = 1; off < 16; off <<= 1) {
#pragma unroll
    for (int v = 0; v < 8; ++v)
      rmin[v] = fminf(rmin[v], __shfl_xor(rmin[v], off, 32));
  }

  // Sum the 8 row-mins held by this half-wave; combine halves (rows 0-7 + 8-15).
  float s = 0.f;
#pragma unroll
  for (int v = 0; v < 8; ++v) s += rmin[v];
  s += __shfl_xor(s, 16, 32);

  if (lane == 0) atomicAdd(acc, s);
}

__global__ void chamfer_finalize(const float* __restrict__ acc,
                                 float* __restrict__ out,
                                 float invBM, float invBN)
{
  // acc[0] = sum over (b, j) of min_i dist  -> mean(dist1)
  // acc[1] = sum over (b, i) of min_j dist  -> mean(dist2)
  out[0] = acc[0] * invBM + acc[1] * invBN;
}

extern "C" void kernel_launch(void* const* d_in, const int* in_sizes, int n_in,
                              void* d_out, int out_size, void* d_ws, size_t ws_size,
                              hipStream_t stream) {
  const float* x = (const float*)d_in[0];   // [B, N, 3]
  const float* y = (const float*)d_in[1];   // [B, M, 3]
  const int B = 8;
  const int N = in_sizes[0] / (B * 3);
  const int M = in_sizes[1] / (B * 3);

  float* acc = (float*)d_ws;                // acc[0], acc[1]
  hipMemsetAsync(d_ws, 0, 2 * sizeof(float), stream);

  // dist1: per-y-point nearest over x  (P = y, Q = x), mean over B*M
  {
    int blocks = (B * (M / TILE)) / WAVES_PER_BLOCK;
    chamfer_rowmin_kernel<<<blocks, BLOCK, 0, stream>>>(y, x, M, N, acc + 0);
  }
  // dist2: per-x-point nearest over y  (P = x, Q = y), mean over B*N
  {
    int blocks = (B * (N / TILE)) / WAVES_PER_BLOCK;
    chamfer_rowmin_kernel<<<blocks, BLOCK, 0, stream>>>(x, y, N, M, acc + 1);
  }

  chamfer_finalize<<<1, 1, 0, stream>>>(acc, (float*)d_out,
                                        1.0f / (float)(B * M),
                                        1.0f / (float)(B * N));
}